// Quantizer_41085657153785
// MI455X (gfx1250) — compile-verified
//
#include <hip/hip_runtime.h>
#include <hip/hip_bf16.h>

// MX-FP4 quantizer for MI455X (gfx1250).
// Memory-bound: ~260 MiB of traffic -> ~11 us at 23.3 TB/s. No matrix math,
// so the CDNA5-specific path used here is the async global->LDS pipeline
// (GLOBAL_LOAD_ASYNC_TO_LDS_B128 + S_WAIT_ASYNCCNT, double-buffered per wave)
// plus non-temporal b128 stores for the streaming output.

typedef __attribute__((ext_vector_type(4))) float fvec4;
typedef __attribute__((ext_vector_type(4))) int   v4i;

#define FP4_MAX 6.0f

#if defined(__AMDGCN__) && __has_builtin(__builtin_amdgcn_global_load_async_to_lds_b128)
#define HAVE_ASYNC_LDS 1
#else
#define HAVE_ASYNC_LDS 0
#endif

// ---- gfx1250 async-copy helpers -------------------------------------------

#if HAVE_ASYNC_LDS
__device__ __forceinline__ void glds_copy16(const float* g, float* l) {
  // 16 bytes per lane, global -> LDS, tracked by ASYNCcnt.
  // Builtin signature (from hipcc diagnostic): (global v4i*, local v4i*, imm, cpol)
  __builtin_amdgcn_global_load_async_to_lds_b128(
      (__attribute__((address_space(1))) v4i*)(const_cast<float*>(g)),
      (__attribute__((address_space(3))) v4i*)(l),
      /*imm offset*/ 0, /*cpol*/ 0);
}
#endif

__device__ __forceinline__ void wait_async_le(int n) {
#if defined(__AMDGCN__)
#if __has_builtin(__builtin_amdgcn_s_wait_asynccnt)
  if (n == 0) __builtin_amdgcn_s_wait_asynccnt(0);
  else        __builtin_amdgcn_s_wait_asynccnt(1);
#else
  if (n == 0) asm volatile("s_wait_asynccnt 0" ::: "memory");
  else        asm volatile("s_wait_asynccnt 1" ::: "memory");
#endif
  asm volatile("" ::: "memory");  // keep LDS reads below the wait
#endif
}

// ---- MX-FP4 math -----------------------------------------------------------

// shared_exp = clip(floor(log2(amax>0 ? amax : 1)) - 2, -127, 127)
// Exponent-field extraction is exact floor(log2) for normals; for denormal
// amax both the true value and this approximation clamp to -127.
__device__ __forceinline__ int shared_exp_i(float amax) {
  const float safe = (amax > 0.0f) ? amax : 1.0f;
  int e = (int)((__float_as_uint(safe) >> 23) & 0xFFu) - 127 - 2;
  if (e < -127) e = -127;
  if (e >  127) e =  127;
  return e;
}

// Round |x|*2^-e (clipped to 6) onto the FP4 E2M1 grid with RNE, reapply sign.
__device__ __forceinline__ float quant_one(float xv, int e) {
  const float a = fminf(ldexpf(fabsf(xv), -e), FP4_MAX);  // exact pow2 scaling
  const float r = (a < 2.0f) ? (rintf(a + a) * 0.5f)
                : (a < 4.0f) ? rintf(a)
                             : fminf(rintf(a * 0.5f) * 2.0f, FP4_MAX);
  return copysignf(r, xv);
}

// One wave-tile: lane owns 4 consecutive elements; 8 lanes = one MX block.
__device__ __forceinline__ void process_tile(const fvec4 v, long long t, int lane,
                                             float* __restrict__ qout,
                                             float* __restrict__ sout) {
  float amax = fmaxf(fmaxf(fabsf(v.x), fabsf(v.y)), fmaxf(fabsf(v.z), fabsf(v.w)));
  amax = fmaxf(amax, __shfl_xor(amax, 1, 32));
  amax = fmaxf(amax, __shfl_xor(amax, 2, 32));
  amax = fmaxf(amax, __shfl_xor(amax, 4, 32));  // 8-lane group amax (wave32)
  const int e = shared_exp_i(amax);

  fvec4 q;
  q.x = quant_one(v.x, e);
  q.y = quant_one(v.y, e);
  q.z = quant_one(v.z, e);
  q.w = quant_one(v.w, e);
  __builtin_nontemporal_store(q, (fvec4*)(qout + (t << 7) + (lane << 2)));

  if ((lane & 7) == 0) {
    sout[(t << 2) + (lane >> 3)] = ldexpf(1.0f, e);
  }
}

// ---- kernel ----------------------------------------------------------------

__global__ void __launch_bounds__(256)
mxfp4_quant_kernel(const float* __restrict__ x, float* __restrict__ qout,
                   float* __restrict__ sout, long long n) {
  const int lane = (int)(threadIdx.x & 31);
  const int wav  = (int)(threadIdx.x >> 5);
  const long long tiles   = n >> 7;  // 128 elements per wave-tile
  const long long w0      = (((long long)blockIdx.x * blockDim.x) + threadIdx.x) >> 5;
  const long long wstride = ((long long)gridDim.x * blockDim.x) >> 5;

#if HAVE_ASYNC_LDS
  // 8 waves x 2 buffers x 128 floats (512 B per wave-tile) = 8 KiB.
  __shared__ __align__(16) float lds[8 * 2 * 128];
  float* my = &lds[wav * 256];

  if (w0 < tiles) {  // prefetch tile 0 into buffer 0 (uniform per wave)
    glds_copy16(x + (w0 << 7) + (lane << 2), my + (lane << 2));
  }
  long long k = 0;
  for (long long t = w0; t < tiles; t += wstride, ++k) {
    const int cur = (int)(k & 1);
    const long long tn = t + wstride;
    if (tn < tiles) {
      glds_copy16(x + (tn << 7) + (lane << 2), my + ((cur ^ 1) * 128) + (lane << 2));
      wait_async_le(1);  // async loads complete in order: tile t is ready
    } else {
      wait_async_le(0);
    }
    const fvec4 v = *(const fvec4*)(my + cur * 128 + (lane << 2));
    process_tile(v, t, lane, qout, sout);
  }
#else
  for (long long t = w0; t < tiles; t += wstride) {
    const fvec4 v = __builtin_nontemporal_load(
        (const fvec4*)(x + (t << 7) + (lane << 2)));
    process_tile(v, t, lane, qout, sout);
  }
#endif

  // Ragged tail: n % 128 leftover elements = up to 3 MX blocks, one thread each.
  const long long done = tiles << 7;
  const long long remBlocks = (n - done) >> 5;
  if (remBlocks > 0) {
    const long long gid = (long long)blockIdx.x * blockDim.x + threadIdx.x;
    if (gid < remBlocks) {
      const long long b = (done >> 5) + gid;
      const float* xb = x + (b << 5);
      float amax = 0.0f;
      for (int i = 0; i < 32; ++i) amax = fmaxf(amax, fabsf(xb[i]));
      const int e = shared_exp_i(amax);
      sout[b] = ldexpf(1.0f, e);
      float* qb = qout + (b << 5);
      for (int i = 0; i < 32; ++i) qb[i] = quant_one(xb[i], e);
    }
  }
}

// ---- launch ----------------------------------------------------------------

extern "C" void kernel_launch(void* const* d_in, const int* in_sizes, int n_in,
                              void* d_out, int out_size, void* d_ws, size_t ws_size,
                              hipStream_t stream) {
  (void)n_in; (void)out_size; (void)d_ws; (void)ws_size;
  const float* x = (const float*)d_in[0];
  const long long n = (long long)in_sizes[0];   // 4096*8192 = 33,554,432
  float* qout = (float*)d_out;                  // [n] quantized grid values
  float* sout = qout + n;                       // [n/32] E8M0 scales (as fp32)

  const long long tiles = n >> 7;               // wave-tiles of 128 elements
  long long blocksNeeded = (tiles + 7) >> 3;    // 8 waves per 256-thread block
  if (blocksNeeded < 1)    blocksNeeded = 1;
  if (blocksNeeded > 4096) blocksNeeded = 4096; // grid-stride covers the rest

  mxfp4_quant_kernel<<<(int)blocksNeeded, 256, 0, stream>>>(x, qout, sout, n);
}